// ConLoss_70918499991901
// MI455X (gfx1250) — compile-verified
//
#include <hip/hip_runtime.h>
#include <hip/hip_bf16.h>

#define D_DIM 128
#define G_NUM 64
#define T_INV (1.0f / 0.07f)

typedef float v2f __attribute__((ext_vector_type(2)));
typedef float v8f __attribute__((ext_vector_type(8)));

// ---------------------------------------------------------------------------
// Kernel 1: group-summed source vectors  gs[g][d] = sum_{s: seg_s[s]==g} zs[s][d]
// (segment_sum is linear, so the "numerator" GEMM collapses to 64 x 128 sums)
// Deterministic: fixed loop order, no atomics.
// ---------------------------------------------------------------------------
__global__ __launch_bounds__(D_DIM) void conloss_group_sums(
    const float* __restrict__ zs, const int* __restrict__ seg_s,
    float* __restrict__ gs, float* __restrict__ counts, int N)
{
    const int g = blockIdx.x;
    const int d = threadIdx.x;
    float acc = 0.f;
    int cnt = 0;
    for (int s = 0; s < N; ++s) {
        if (seg_s[s] == g) { acc += zs[(size_t)s * D_DIM + d]; ++cnt; }
    }
    gs[g * D_DIM + d] = acc;
    if (d == 0) counts[g] = (float)cnt;
}

// ---------------------------------------------------------------------------
// Kernel 2: fused GEMM + exp + column-reduce using V_WMMA_F32_16X16X4_F32.
// One block (8 waves) per 16-column tile of Zt. Each wave strides over
// 16-row blocks of concat(Zt, Zs), computing 16x16 f32 sims tiles with
// 32 chained fp32 WMMAs (K=128 in steps of 4), then exp+accumulate.
// ---------------------------------------------------------------------------
__global__ __launch_bounds__(256) void conloss_den_wmma(
    const float* __restrict__ zt, const float* __restrict__ zs,
    const int* __restrict__ seg_t, const float* __restrict__ gs,
    const float* __restrict__ counts, float* __restrict__ term,
    int M, int N)
{
    __shared__ float red[8 * 16];
    const int lane = threadIdx.x & 31;
    const int wave = threadIdx.x >> 5;   // 0..7
    const int half = lane >> 4;          // 0 or 1
    const int lc   = lane & 15;
    const int col  = blockIdx.x * 16 + lc;

    // B fragment (column 'col' of Zt^T == row 'col' of Zt), resident in VGPRs.
    // ISA layout: lane L -> N = L&15; VGPR pair holds K = 4k + 2*half, +1.
    v2f bfrag[32];
    {
        const float* brow = zt + (size_t)col * D_DIM;
        #pragma unroll
        for (int k = 0; k < 32; ++k)
            bfrag[k] = *(const v2f*)(brow + 4 * k + 2 * half);
    }

    const int RB = (M + N) >> 4;   // 16-row blocks over concat(Zt, Zs)
    float den_acc = 0.f;
    for (int rb = wave; rb < RB; rb += 8) {
        const int r0 = rb * 16 + lc;
        const float* arow = (r0 < M) ? (zt + (size_t)r0 * D_DIM)
                                     : (zs + (size_t)(r0 - M) * D_DIM);
        // Prefetch this wave's next row block (emits global_prefetch_b8).
        const int rn = r0 + 8 * 16;
        if (rn < M + N) {
            const float* pn = (rn < M) ? (zt + (size_t)rn * D_DIM)
                                       : (zs + (size_t)(rn - M) * D_DIM);
            __builtin_prefetch(pn, 0, 1);
        }

        v8f c = {};
        #pragma unroll
        for (int k = 0; k < 32; ++k) {
            v2f a = *(const v2f*)(arow + 4 * k + 2 * half);
            // 8 args: (neg_a, A, neg_b, B, c_mod, C, reuse_a, reuse_b)
            c = __builtin_amdgcn_wmma_f32_16x16x4_f32(
                    false, a, false, bfrag[k], (short)0, c, false, false);
        }
        // C layout: lanes 0-15 hold rows 0..7 of column lc (VGPR 0..7),
        // lanes 16-31 hold rows 8..15 of the same column.
        #pragma unroll
        for (int r = 0; r < 8; ++r)
            den_acc += expf(c[r] * T_INV);
    }

    // Combine the two row-halves of each column, then reduce across waves.
    den_acc += __shfl_xor(den_acc, 16, 32);
    if (lane < 16) red[wave * 16 + lc] = den_acc;
    __syncthreads();

    if (threadIdx.x < 16) {
        const int m = blockIdx.x * 16 + (int)threadIdx.x;
        float den = 0.f;
        #pragma unroll
        for (int w = 0; w < 8; ++w) den += red[w * 16 + threadIdx.x];

        const float* zm = zt + (size_t)m * D_DIM;
        const int    g  = seg_t[m];
        const float* gv = gs + g * D_DIM;
        float self = 0.f, num = 0.f;
        for (int d = 0; d < D_DIM; ++d) {
            const float v = zm[d];
            self += v * v;       // self-similarity (exact, not from GEMM tile)
            num  += gv[d] * v;   // group-sum dot -> mean numerator
        }
        den -= expf(self * T_INV);
        term[m] = logf(den) - (num * T_INV) / counts[g];
    }
}

// ---------------------------------------------------------------------------
// Kernel 3: deterministic tree reduction of the M per-column terms.
// ---------------------------------------------------------------------------
__global__ __launch_bounds__(1024) void conloss_reduce(
    const float* __restrict__ term, float* __restrict__ out, int M)
{
    __shared__ float s[1024];
    float acc = 0.f;
    for (int i = threadIdx.x; i < M; i += 1024) acc += term[i];
    s[threadIdx.x] = acc;
    __syncthreads();
    for (int st = 512; st > 0; st >>= 1) {
        if ((int)threadIdx.x < st) s[threadIdx.x] += s[threadIdx.x + st];
        __syncthreads();
    }
    if (threadIdx.x == 0) out[0] = s[0];
}

// ---------------------------------------------------------------------------
extern "C" void kernel_launch(void* const* d_in, const int* in_sizes, int n_in,
                              void* d_out, int out_size, void* d_ws, size_t ws_size,
                              hipStream_t stream) {
    (void)n_in; (void)out_size; (void)ws_size;
    const float* zs    = (const float*)d_in[0];   // z_source  [N, 128]
    const float* zt    = (const float*)d_in[1];   // z_target  [M, 128]
    const int*   seg_s = (const int*)d_in[2];     // seg_source [N]
    const int*   seg_t = (const int*)d_in[3];     // seg_target [M]
    const int N = in_sizes[0] / D_DIM;
    const int M = in_sizes[1] / D_DIM;

    float* gs     = (float*)d_ws;              // G_NUM * D_DIM
    float* counts = gs + G_NUM * D_DIM;        // G_NUM
    float* term   = counts + G_NUM;            // M

    conloss_group_sums<<<G_NUM, D_DIM, 0, stream>>>(zs, seg_s, gs, counts, N);
    conloss_den_wmma<<<M / 16, 256, 0, stream>>>(zt, zs, seg_t, gs, counts, term, M, N);
    conloss_reduce<<<1, 1024, 0, stream>>>(term, (float*)d_out, M);
}